// LSTMModelDefinition_48284022342118
// MI455X (gfx1250) — compile-verified
//
#include <hip/hip_runtime.h>

typedef float v2f __attribute__((ext_vector_type(2)));
typedef float v8f __attribute__((ext_vector_type(8)));

namespace {
constexpr int kB         = 4096;
constexpr int kT         = 512;
constexpr int kH         = 32;    // hidden size
constexpr int kK         = 64;    // concat K = [input(32) | hidden(32)]
constexpr int kKSteps    = kK / 4;
constexpr int kStr       = 68;    // padded LDS row stride (dwords): bank skew, 16B aligned
constexpr int kRowsPerWG = 16;    // one 16-row batch tile per WG, 2 waves cooperate
}

struct SMem {
    float S0[kRowsPerWG * kStr];  // layer0 A tile: [relu(x*w1+b1) | h0]
    float S1[kRowsPerWG * kStr];  // layer1 A tile: [h0 | h1]
};

#if __has_builtin(__builtin_amdgcn_tanhf)
// CDNA5 hardware transcendental V_TANH_F32: 1 trans op per activation.
__device__ __forceinline__ float tanh_f(float x) { return __builtin_amdgcn_tanhf(x); }
#else
__device__ __forceinline__ float tanh_f(float x) {
    float e = __expf(-2.0f * x);
    return (1.0f - e) / (1.0f + e);
}
#endif

// Half of a 16x128 gate block: wave p computes tiles nt = 2g+p, i.e. gates
// i,f,g,o for column block j=p. 64 WMMAs. Split-K parity accumulators give
// 8 independent chains (even ks -> accE, odd ks -> accO) so same-chain WMMAs
// are 8 issues apart, covering the XDL dependency latency; merged with one
// packed-FP32 vector add per gate at the end.
__device__ __forceinline__ void gemm4(const float* Sbuf,
                                      const v2f (&wf)[4][kKSteps],
                                      int idx, int hf, const v8f& zacc,
                                      v8f acc[4]) {
    v8f accE[4], accO[4];
    {
        // A 16x4 fp32 layout: lane = 16*hf + m holds A[m][k0+2hf], A[m][k0+2hf+1]
        v2f a0 = *(const v2f*)(Sbuf + idx * kStr + 0 + 2 * hf);
        v2f a1 = *(const v2f*)(Sbuf + idx * kStr + 4 + 2 * hf);
#pragma unroll
        for (int g = 0; g < 4; ++g)
            accE[g] = __builtin_amdgcn_wmma_f32_16x16x4_f32(
                false, a0, false, wf[g][0], (short)0, zacc, false, false);
#pragma unroll
        for (int g = 0; g < 4; ++g)
            accO[g] = __builtin_amdgcn_wmma_f32_16x16x4_f32(
                false, a1, false, wf[g][1], (short)0, zacc, false, false);
    }
#pragma unroll
    for (int ks = 2; ks < kKSteps; ks += 2) {
        v2f a0 = *(const v2f*)(Sbuf + idx * kStr + 4 * ks + 2 * hf);
        v2f a1 = *(const v2f*)(Sbuf + idx * kStr + 4 * (ks + 1) + 2 * hf);
#pragma unroll
        for (int g = 0; g < 4; ++g)
            accE[g] = __builtin_amdgcn_wmma_f32_16x16x4_f32(
                false, a0, false, wf[g][ks], (short)0, accE[g], false, false);
#pragma unroll
        for (int g = 0; g < 4; ++g)
            accO[g] = __builtin_amdgcn_wmma_f32_16x16x4_f32(
                false, a1, false, wf[g][ks + 1], (short)0, accO[g], false, false);
    }
#pragma unroll
    for (int g = 0; g < 4; ++g) acc[g] = accE[g] + accO[g];
}

// Cell update for this wave's 16 gate columns. sigmoid(x) = 0.5*tanh(0.5x)+0.5
// with the 0.5*bias prefolded -> each sigmoid = fma + v_tanh + fma.
// h store to LDS = D-layout -> A-layout transpose for the next matmul.
__device__ __forceinline__ void update4(const v8f acc[4], const float (&hb)[3],
                                        float bg, v8f& c, float* dst_in,
                                        float* dst_rec, int col, int hf) {
#pragma unroll
    for (int r = 0; r < 8; ++r) {
        float iv = fmaf(0.5f, tanh_f(fmaf(0.5f, acc[0][r], hb[0])), 0.5f);
        float fv = fmaf(0.5f, tanh_f(fmaf(0.5f, acc[1][r], hb[1])), 0.5f);
        float gv = tanh_f(acc[2][r] + bg);
        float ov = fmaf(0.5f, tanh_f(fmaf(0.5f, acc[3][r], hb[2])), 0.5f);
        float cv = fv * c[r] + iv * gv;
        c[r] = cv;
        float hv = ov * tanh_f(cv);
        int row = r + 8 * hf;                       // D layout: vgpr r -> rows r / r+8
        if (dst_in) dst_in[row * kStr + col] = hv;  // layer-(l+1) input slot
        dst_rec[row * kStr + kH + col] = hv;        // own recurrent slot
    }
}

__global__ __launch_bounds__(64) void lstm_fused_kernel(
    const float* __restrict__ X,     // [B,T,1]
    const float* __restrict__ w1, const float* __restrict__ b1,
    const float* __restrict__ Wih0, const float* __restrict__ Whh0,
    const float* __restrict__ bih0, const float* __restrict__ bhh0,
    const float* __restrict__ Wih1, const float* __restrict__ Whh1,
    const float* __restrict__ bih1, const float* __restrict__ bhh1,
    const float* __restrict__ w2, const float* __restrict__ b2,
    float* __restrict__ out) {
    __shared__ SMem sm;
    const int tid  = threadIdx.x;
    const int p    = tid >> 5;    // wave id: owns gate-column block j = p
    const int lane = tid & 31;
    const int idx  = lane & 15;   // M for A-frags / N for B-frags & D cols
    const int hf   = lane >> 4;
    const int row_base = blockIdx.x * kRowsPerWG;

    // h starts at zero -> zero both staging tiles (S0,S1 contiguous in struct)
    for (int i = tid; i < 2 * kRowsPerWG * kStr; i += 64) sm.S0[i] = 0.0f;

    // ---- per-wave weight B-fragments straight from global into registers ----
    // B 4x16 fp32 layout: lane = 16*hf + n holds W[n][k0+2hf], W[n][k0+2hf+1].
    // Wave p's tile nt = 2g+p -> weight row grow = (2g+p)*16+idx. k<32 -> Wih,
    // k>=32 -> Whh (the concat K). One-time; all 256 WGs hit the same L2 lines.
    v2f w0f[4][kKSteps], w1f[4][kKSteps];
#pragma unroll
    for (int g = 0; g < 4; ++g) {
        int grow = (2 * g + p) * 16 + idx;
#pragma unroll
        for (int ks = 0; ks < kKSteps; ++ks) {
            int k0 = 4 * ks + 2 * hf;   // ks<8 -> Wih half, ks>=8 -> Whh half
            w0f[g][ks] = (ks < 8) ? *(const v2f*)(Wih0 + grow * kH + k0)
                                  : *(const v2f*)(Whh0 + grow * kH + (k0 - kH));
            w1f[g][ks] = (ks < 8) ? *(const v2f*)(Wih1 + grow * kH + k0)
                                  : *(const v2f*)(Whh1 + grow * kH + (k0 - kH));
        }
    }

    // Per-lane biases for this wave's gate column (i,f,o prefolded as 0.5*b
    // for the tanh-form sigmoid; g gate keeps the full bias).
    const int col = p * 16 + idx;        // column within each 32-wide gate block
    float hb0[3], hb1[3], bg0, bg1;
    hb0[0] = 0.5f * (bih0[0 * kH + col] + bhh0[0 * kH + col]);
    hb0[1] = 0.5f * (bih0[1 * kH + col] + bhh0[1 * kH + col]);
    bg0    =        bih0[2 * kH + col] + bhh0[2 * kH + col];
    hb0[2] = 0.5f * (bih0[3 * kH + col] + bhh0[3 * kH + col]);
    hb1[0] = 0.5f * (bih1[0 * kH + col] + bhh1[0 * kH + col]);
    hb1[1] = 0.5f * (bih1[1 * kH + col] + bhh1[1 * kH + col]);
    bg1    =        bih1[2 * kH + col] + bhh1[2 * kH + col];
    hb1[2] = 0.5f * (bih1[3 * kH + col] + bhh1[3 * kH + col]);

    // Layer-0 input stage assignment: 64 threads = 16 rows x 4 col-groups of 8.
    const int arow = tid & 15;
    const int acg  = tid >> 4;           // 0..3 -> cols [8*acg, 8*acg+8)
    float w1r[8], b1r[8];
#pragma unroll
    for (int e = 0; e < 8; ++e) { w1r[e] = w1[8 * acg + e]; b1r[e] = b1[8 * acg + e]; }
    const float* xrow = X + (size_t)(row_base + arow) * kT;

    const v8f zacc = {0.f, 0.f, 0.f, 0.f, 0.f, 0.f, 0.f, 0.f};
    v8f c0 = zacc, c1 = zacc, acc[4];

    __syncthreads();   // staging zeroed, both waves ready

    for (int t = 0; t < kT; ++t) {
        // a[row][col] = relu(x[row,t]*w1[col]+b1[col]) into S0[:,0:32]
        float xv = xrow[t];
#pragma unroll
        for (int q = 0; q < 2; ++q) {
            int cc = 8 * acg + 4 * q;
            float4 av;
            av.x = fmaxf(fmaf(xv, w1r[4 * q + 0], b1r[4 * q + 0]), 0.0f);
            av.y = fmaxf(fmaf(xv, w1r[4 * q + 1], b1r[4 * q + 1]), 0.0f);
            av.z = fmaxf(fmaf(xv, w1r[4 * q + 2], b1r[4 * q + 2]), 0.0f);
            av.w = fmaxf(fmaf(xv, w1r[4 * q + 3], b1r[4 * q + 3]), 0.0f);
            *(float4*)(sm.S0 + arow * kStr + cc) = av;
        }
        __syncthreads();
        gemm4(sm.S0, w0f, idx, hf, zacc, acc);              // layer 0, this wave's gates
        __syncthreads();
        update4(acc, hb0, bg0, c0, sm.S1, sm.S0, col, hf);  // h0 -> S1 in + S0 rec
        __syncthreads();
        gemm4(sm.S1, w1f, idx, hf, zacc, acc);              // layer 1, this wave's gates
        __syncthreads();
        update4(acc, hb1, bg1, c1, nullptr, sm.S1, col, hf);// h1 -> S1 rec
    }
    __syncthreads();

    // out[b] = [h0_final | h1_final] . w2 + b2   (S1 holds exactly that concat)
    if (tid < 16) {
        float s = b2[0];
#pragma unroll
        for (int k = 0; k < 2 * kH; ++k) s += w2[k] * sm.S1[tid * kStr + k];
        out[row_base + tid] = s;
    }
}

extern "C" void kernel_launch(void* const* d_in, const int* in_sizes, int n_in,
                              void* d_out, int out_size, void* d_ws, size_t ws_size,
                              hipStream_t stream) {
    const float* X    = (const float*)d_in[0];
    const float* w1   = (const float*)d_in[1];
    const float* b1   = (const float*)d_in[2];
    const float* Wih0 = (const float*)d_in[3];
    const float* Whh0 = (const float*)d_in[4];
    const float* bih0 = (const float*)d_in[5];
    const float* bhh0 = (const float*)d_in[6];
    const float* Wih1 = (const float*)d_in[7];
    const float* Whh1 = (const float*)d_in[8];
    const float* bih1 = (const float*)d_in[9];
    const float* bhh1 = (const float*)d_in[10];
    const float* w2   = (const float*)d_in[11];
    const float* b2   = (const float*)d_in[12];
    float* out = (float*)d_out;

    dim3 grid(kB / kRowsPerWG);   // 256 WGs
    dim3 block(64);               // 2 cooperating wave32s
    lstm_fused_kernel<<<grid, block, 0, stream>>>(
        X, w1, b1, Wih0, Whh0, bih0, bhh0, Wih1, Whh1, bih1, bhh1, w2, b2, out);
}